// LSTMExtractor_90065464197420
// MI455X (gfx1250) — compile-verified
//
#include <hip/hip_runtime.h>

typedef float v2f __attribute__((ext_vector_type(2)));
typedef float v8f __attribute__((ext_vector_type(8)));

#define BB 2048
#define TT 512
#define FF 60
#define HH 7
#define G4 28   // 4*H gates

// sigmoid(x) = 1/(1+2^(-x*log2e))  -> v_exp_f32 + v_rcp_f32
__device__ __forceinline__ float sigm_fast(float x) {
    return __builtin_amdgcn_rcpf(1.0f + __builtin_amdgcn_exp2f(-1.44269504f * x));
}
// tanh(x) = 2*sigmoid(2x) - 1
__device__ __forceinline__ float tanh_fast(float x) {
    return __builtin_amdgcn_rcpf(1.0f + __builtin_amdgcn_exp2f(-2.88539008f * x)) * 2.0f - 1.0f;
}

// ---------------- Phase 1: gx = x @ Wih0^T + (bih0+bhh0), via V_WMMA_F32_16X16X4_F32 ----
// One wave computes a 16(M) x 32(N-padded) tile of gx; K=60 in 15 steps of 4.
// ISA 32-bit A 16x4 layout: lanes 0-15 hold M=lane, VGPR0=K0,VGPR1=K1; lanes 16-31: K2,K3.
__global__ __launch_bounds__(256) void lstm_xproj_wmma(
    const float* __restrict__ x, const float* __restrict__ Wih0,
    const float* __restrict__ bih0, const float* __restrict__ bhh0,
    float* __restrict__ gx) {
    const int lane  = threadIdx.x & 31;
    const int wave  = threadIdx.x >> 5;
    const int mtile = blockIdx.x * 8 + wave;
    const int half  = lane >> 4;          // 0: K+0..1, 1: K+2..3
    const int l15   = lane & 15;
    const int row   = mtile * 16 + l15;   // A-matrix row fed by this lane
    const int n0    = l15;                // B col tile 0: gates 0..15
    const int n1    = 16 + l15;           // B col tile 1: gates 16..31 (28.. padded)
    const int n1c   = (n1 < G4) ? n1 : (G4 - 1);       // clamped, branch-free
    const float m1  = (n1 < G4) ? 1.0f : 0.0f;         // zero mask for pad columns

    v8f acc0 = {};
    v8f acc1 = {};
    #pragma unroll
    for (int k0 = 0; k0 < FF; k0 += 4) {
        const int kb = k0 + half * 2;
        v2f a;  a.x  = x[row * FF + kb];        a.y  = x[row * FF + kb + 1];
        v2f b0; b0.x = Wih0[n0  * FF + kb];     b0.y = Wih0[n0  * FF + kb + 1];
        v2f b1; b1.x = Wih0[n1c * FF + kb]*m1;  b1.y = Wih0[n1c * FF + kb + 1]*m1;
        acc0 = __builtin_amdgcn_wmma_f32_16x16x4_f32(false, a, false, b0, (short)0, acc0, false, false);
        acc1 = __builtin_amdgcn_wmma_f32_16x16x4_f32(false, a, false, b1, (short)0, acc1, false, false);
    }
    const float bias0 = bih0[n0] + bhh0[n0];
    const float bias1 = (bih0[n1c] + bhh0[n1c]) * m1;
    // C/D layout: VGPR v -> row (v + half*8), col (lane&15) per tile
    #pragma unroll
    for (int v = 0; v < 8; ++v) {
        const int rm = mtile * 16 + v + half * 8;
        gx[(size_t)rm * G4 + n0] = acc0[v] + bias0;
        if (n1 < G4) gx[(size_t)rm * G4 + n1] = acc1[v] + bias1;
    }
}

// ---------------- Phase 2: sequential recurrence, one wave32 per batch row --------------
// Lane g in [0,28) owns gate g of both layers; hidden state replicated across lanes.
__global__ __launch_bounds__(256) void lstm_recur(
    const float* __restrict__ gx,
    const float* __restrict__ Whh0, const float* __restrict__ Wih1,
    const float* __restrict__ Whh1, const float* __restrict__ bih1,
    const float* __restrict__ bhh1, float* __restrict__ out) {
    const int lane = threadIdx.x & 31;
    const int wave = threadIdx.x >> 5;
    const int b    = blockIdx.x * 8 + wave;
    const int g    = (lane < G4) ? lane : (G4 - 1);

    float w0[HH], wi1[HH], wh1[HH];
    #pragma unroll
    for (int j = 0; j < HH; ++j) {
        w0[j]  = Whh0[g * HH + j];
        wi1[j] = Wih1[g * HH + j];
        wh1[j] = Whh1[g * HH + j];
    }
    const float b1g  = bih1[g] + bhh1[g];
    const bool  isg  = (lane >= 14 && lane < 21);   // 'g' gates use tanh
    const float scl  = isg ? 2.0f : 1.0f;
    const float mulA = isg ? 2.0f : 1.0f;
    const float addB = isg ? -1.0f : 0.0f;
    const int   jj   = lane % HH;                   // owner gather index

    float h0r[HH] = {}, h1r[HH] = {};
    float c0o = 0.0f, c1o = 0.0f, h1own = 0.0f;
    const float* __restrict__ gxr = gx + (size_t)b * TT * G4;

    for (int t = 0; t < TT; ++t) {
        __builtin_prefetch((const void*)(gxr + (t + 6) * G4 + g), 0, 1);
        // ---- layer 0 gate ----
        float pre = gxr[t * G4 + g];                // bias+x-projection precomputed
        #pragma unroll
        for (int j = 0; j < HH; ++j) pre = fmaf(w0[j], h0r[j], pre);
        float a = fmaf(sigm_fast(pre * scl), mulA, addB);
        // owner lanes (lane<7) gather i,f,g,o for hidden unit jj
        float gi = __shfl(a, jj);
        float gf = __shfl(a, jj + 7);
        float gg = __shfl(a, jj + 14);
        float go = __shfl(a, jj + 21);
        float c0n = fmaf(gf, c0o, gi * gg);
        c0o = c0n;
        float h0n = go * tanh_fast(c0n);
        #pragma unroll
        for (int j = 0; j < HH; ++j) h0r[j] = __shfl(h0n, j);   // re-replicate h0
        // ---- layer 1 gate ----
        float pre1 = b1g;
        #pragma unroll
        for (int j = 0; j < HH; ++j) pre1 = fmaf(wi1[j], h0r[j], pre1);
        #pragma unroll
        for (int j = 0; j < HH; ++j) pre1 = fmaf(wh1[j], h1r[j], pre1);
        float a1 = fmaf(sigm_fast(pre1 * scl), mulA, addB);
        float gi1 = __shfl(a1, jj);
        float gf1 = __shfl(a1, jj + 7);
        float gg1 = __shfl(a1, jj + 14);
        float go1 = __shfl(a1, jj + 21);
        float c1n = fmaf(gf1, c1o, gi1 * gg1);
        c1o = c1n;
        h1own = go1 * tanh_fast(c1n);
        #pragma unroll
        for (int j = 0; j < HH; ++j) h1r[j] = __shfl(h1own, j); // re-replicate h1
    }
    if (lane < HH) out[(size_t)b * HH + lane] = h1own;
}

extern "C" void kernel_launch(void* const* d_in, const int* in_sizes, int n_in,
                              void* d_out, int out_size, void* d_ws, size_t ws_size,
                              hipStream_t stream) {
    const float* x    = (const float*)d_in[0];
    const float* Wih0 = (const float*)d_in[1];
    const float* Whh0 = (const float*)d_in[2];
    const float* bih0 = (const float*)d_in[3];
    const float* bhh0 = (const float*)d_in[4];
    // d_in[5] = Wih1, d_in[6] = Whh1, d_in[7] = bih1, d_in[8] = bhh1
    const float* Wih1 = (const float*)d_in[5];
    const float* Whh1 = (const float*)d_in[6];
    const float* bih1 = (const float*)d_in[7];
    const float* bhh1 = (const float*)d_in[8];
    float* out = (float*)d_out;
    float* gx  = (float*)d_ws;   // needs B*T*28*4 = 112 MiB of scratch

    const int mtiles = (BB * TT) / 16;             // 65536 M-tiles
    lstm_xproj_wmma<<<mtiles / 8, 256, 0, stream>>>(x, Wih0, bih0, bhh0, gx);
    lstm_recur<<<BB / 8, 256, 0, stream>>>(gx, Whh0, Wih1, Whh1, bih1, bhh1, out);
}